// MultiTimeAttention_64819646431829
// MI455X (gfx1250) — compile-verified
//
#include <hip/hip_runtime.h>

// ---- problem constants (match reference) ----
#define B_   16
#define TQ   128
#define T_   128
#define DT   128
#define H_   8
#define DQK  64
#define DV   64
#define HD   512   // H_*DQK

typedef __attribute__((ext_vector_type(2))) float v2f;
typedef __attribute__((ext_vector_type(8))) float v8f;

// ---------------------------------------------------------------------------
// Wave-level 16x16 fp32 tile GEMM using V_WMMA_F32_16X16X4_F32.
// ISA layout (7.12.2): A 16x4 -> lanes 0-15: row=lane, K={0,1} in v0,v1;
// lanes 16-31: K={2,3}.  B 4x16 mirrored (col=lane&15).  C/D: VGPR i holds
// row i (lanes 0-15) / row i+8 (lanes 16-31), col = lane&15.
// ---------------------------------------------------------------------------
__device__ __forceinline__ v8f wmma_f32_AB(const float* __restrict__ A, int lda,
                                           const float* __restrict__ Bm, int ldb,
                                           int K, int lane) {
  const int r  = lane & 15;          // A row / B col within tile
  const int kb = (lane >> 4) << 1;   // K sub-offset per half-wave
  v8f acc = {};
#pragma unroll 8
  for (int k = 0; k < K; k += 4) {
    v2f a, b;
    a[0] = A[(size_t)r * lda + (k + kb)];
    a[1] = A[(size_t)r * lda + (k + kb + 1)];
    b[0] = Bm[(size_t)(k + kb) * ldb + r];
    b[1] = Bm[(size_t)(k + kb + 1) * ldb + r];
    acc = __builtin_amdgcn_wmma_f32_16x16x4_f32(false, a, false, b,
                                                (short)0, acc, false, false);
  }
  return acc;
}

// Same, but B given transposed (Bt is N x K row-major): computes A @ Bt^T.
__device__ __forceinline__ v8f wmma_f32_ABt(const float* __restrict__ A, int lda,
                                            const float* __restrict__ Bt, int ldb,
                                            int K, int lane) {
  const int r  = lane & 15;
  const int kb = (lane >> 4) << 1;
  v8f acc = {};
#pragma unroll 8
  for (int k = 0; k < K; k += 4) {
    v2f a, b;
    a[0] = A[(size_t)r * lda + (k + kb)];
    a[1] = A[(size_t)r * lda + (k + kb + 1)];
    b[0] = Bt[(size_t)r * ldb + (k + kb)];
    b[1] = Bt[(size_t)r * ldb + (k + kb + 1)];
    acc = __builtin_amdgcn_wmma_f32_16x16x4_f32(false, a, false, b,
                                                (short)0, acc, false, false);
  }
  return acc;
}

__device__ __forceinline__ void store_tile(float* __restrict__ C, int ldc,
                                           v8f acc, int lane, float scale) {
  const int col = lane & 15;
  const int rb  = (lane >> 4) << 3;
#pragma unroll
  for (int i = 0; i < 8; ++i)
    C[(size_t)(rb + i) * ldc + col] = acc[i] * scale;
}

// ---------------------------------------------------------------------------
// Kernel 1: per-batch projection  Y(128x512) = X(128x128) @ W(128x512)
// grid = (B, rows/16, cols/128); block = 256 (8 waves, one 16-col tile each)
// ---------------------------------------------------------------------------
__global__ __launch_bounds__(256)
void mtan_proj_kernel(const float* __restrict__ X, const float* __restrict__ W,
                      float* __restrict__ Y) {
  const int b    = blockIdx.x;
  const int row0 = blockIdx.y * 16;
  const int wave = threadIdx.x >> 5;
  const int lane = threadIdx.x & 31;
  const int col0 = blockIdx.z * 128 + wave * 16;

  const float* A  = X + ((size_t)b * TQ + row0) * DT;
  const float* Bm = W + col0;
  v8f acc = wmma_f32_AB(A, DT, Bm, HD, DT, lane);

  float* C = Y + ((size_t)b * TQ + row0) * HD + col0;
  store_tile(C, HD, acc, lane, 1.0f);
}

// ---------------------------------------------------------------------------
// Kernel 2: per-(b,h) scores  S(128x128) = Qh(128x64) @ Kh(128x64)^T * 0.125
// grid = (B*H, rows/16); block = 256 (8 waves cover all 128 cols)
// ---------------------------------------------------------------------------
__global__ __launch_bounds__(256)
void mtan_scores_kernel(const float* __restrict__ qh, const float* __restrict__ kh,
                        float* __restrict__ S) {
  const int bh   = blockIdx.x;
  const int b    = bh / H_;
  const int h    = bh % H_;
  const int row0 = blockIdx.y * 16;
  const int wave = threadIdx.x >> 5;
  const int lane = threadIdx.x & 31;
  const int col0 = wave * 16;

  const float* A  = qh + ((size_t)b * TQ + row0) * HD + h * DQK;
  const float* Bt = kh + ((size_t)b * T_ + col0) * HD + h * DQK;
  v8f acc = wmma_f32_ABt(A, HD, Bt, HD, DQK, lane);

  float* C = S + ((size_t)bh * TQ + row0) * T_ + col0;
  store_tile(C, T_, acc, lane, 0.125f);   // 1/sqrt(64)
}

// ---------------------------------------------------------------------------
// Kernel 3 (bandwidth-critical): fused mask + softmax-over-t + sum_t attn*v.
// One 64-thread block per (b,h,q); thread = feature d. Scores row in LDS.
// attn stores are 64 contiguous floats per t-iteration (coalesced 256B) and
// use NON-TEMPORAL hint: the 512MB attn tensor is write-once, never re-read
// on device, so keep it out of WGP$/L2 and preserve cache residency for the
// reused mask/v/scores slices (ISA §7.3 store TH=NT).
// ---------------------------------------------------------------------------
__global__ __launch_bounds__(64)
void mtan_softmax_wsum_kernel(const float* __restrict__ S,
                              const float* __restrict__ mask,
                              const float* __restrict__ V,
                              float* __restrict__ attn,
                              float* __restrict__ outh) {
  const int qi = blockIdx.x;
  const int h  = blockIdx.y;
  const int b  = blockIdx.z;
  const int d  = threadIdx.x;   // 0..63

  __shared__ float s[T_];
  const float* srow = S + ((size_t)(b * H_ + h) * TQ + qi) * T_;
  s[d]      = srow[d];
  s[d + 64] = srow[d + 64];
  __syncthreads();

  const float* mcol = mask + (size_t)b * T_ * DV + d;   // stride DV over t

  // pass 1: masked max
  float m = -3.402823466e38f;
#pragma unroll 4
  for (int t = 0; t < T_; ++t) {
    float val = s[t] - (1.0f - mcol[(size_t)t * DV]) * 1e9f;
    m = fmaxf(m, val);
  }
  // pass 2: denominator
  float sum = 0.0f;
#pragma unroll 4
  for (int t = 0; t < T_; ++t) {
    float val = s[t] - (1.0f - mcol[(size_t)t * DV]) * 1e9f;
    sum += __expf(val - m);
  }
  const float inv = 1.0f / sum;

  // pass 3: stream attn out once (non-temporal), accumulate weighted sum
  float* arow = attn + (((size_t)(b * H_ + h) * TQ + qi) * T_) * DV + d;
  const float* vcol = V + (size_t)b * T_ * DV + d;
  float acc = 0.0f;
#pragma unroll 4
  for (int t = 0; t < T_; ++t) {
    float val = s[t] - (1.0f - mcol[(size_t)t * DV]) * 1e9f;
    float e = __expf(val - m) * inv;
    __builtin_nontemporal_store(e, &arow[(size_t)t * DV]);
    acc += e * vcol[(size_t)t * DV];
  }
  // heads-merged layout (b, q, h*64+d) so final GEMM reads contiguous rows
  outh[((size_t)b * TQ + qi) * HD + h * DV + d] = acc;
}

// ---------------------------------------------------------------------------
// Kernel 4: per-batch final  F(128x128) = Oh(128x512) @ Wo(512x128)
// ---------------------------------------------------------------------------
__global__ __launch_bounds__(256)
void mtan_final_kernel(const float* __restrict__ X, const float* __restrict__ Wo,
                       float* __restrict__ Y) {
  const int b    = blockIdx.x;
  const int row0 = blockIdx.y * 16;
  const int wave = threadIdx.x >> 5;
  const int lane = threadIdx.x & 31;
  const int col0 = wave * 16;

  const float* A  = X + ((size_t)b * TQ + row0) * HD;
  const float* Bm = Wo + col0;
  v8f acc = wmma_f32_AB(A, HD, Bm, 128, HD, lane);

  float* C = Y + ((size_t)b * TQ + row0) * 128 + col0;
  store_tile(C, 128, acc, lane, 1.0f);
}

// ---------------------------------------------------------------------------
extern "C" void kernel_launch(void* const* d_in, const int* in_sizes, int n_in,
                              void* d_out, int out_size, void* d_ws, size_t ws_size,
                              hipStream_t stream) {
  const float* q    = (const float*)d_in[0];
  const float* k    = (const float*)d_in[1];
  const float* v    = (const float*)d_in[2];
  const float* mask = (const float*)d_in[3];
  const float* Wq   = (const float*)d_in[4];
  const float* Wk   = (const float*)d_in[5];
  const float* Wo   = (const float*)d_in[6];

  float* out_final = (float*)d_out;                              // B*TQ*128
  float* out_attn  = out_final + (size_t)B_ * TQ * 128;          // B*H*TQ*T*DV

  // workspace carve-up (20 MB total)
  float* qh     = (float*)d_ws;                                  // B*TQ*HD
  float* kh     = qh     + (size_t)B_ * TQ * HD;                 // B*T*HD
  float* scores = kh     + (size_t)B_ * T_ * HD;                 // B*H*TQ*T
  float* outh   = scores + (size_t)B_ * H_ * TQ * T_;            // B*TQ*HD

  const dim3 blk256(256);
  mtan_proj_kernel<<<dim3(B_, TQ / 16, HD / 128), blk256, 0, stream>>>(q, Wq, qh);
  mtan_proj_kernel<<<dim3(B_, T_ / 16, HD / 128), blk256, 0, stream>>>(k, Wk, kh);
  mtan_scores_kernel<<<dim3(B_ * H_, TQ / 16), blk256, 0, stream>>>(qh, kh, scores);
  mtan_softmax_wsum_kernel<<<dim3(TQ, H_, B_), dim3(64), 0, stream>>>(
      scores, mask, v, out_attn, outh);
  mtan_final_kernel<<<dim3(B_, TQ / 16), blk256, 0, stream>>>(outh, Wo, out_final);
}